// ItemRegressionModel_76733885710730
// MI455X (gfx1250) — compile-verified
//
#include <hip/hip_runtime.h>

// ItemRegressionModel: out[b] = bu + bt + (1/32) * sum_k w[qtu_k, t] * (r[u, qtu_k] - bu - b_item[qtu_k])
// U = I = 2000, K = 32, B = 16384. Pure gather-reduce; all hot tables fit in 192MB L2.

typedef __attribute__((ext_vector_type(2))) float v2f;
typedef __attribute__((ext_vector_type(8))) float v8f;

#define I_DIM 2000
#define K_DIM 32
#define B_DIM 16384
#define OUTS_PER_WAVE 16          // one 16x16 WMMA tile of outputs per wave
#define WAVES_PER_BLOCK 4
#define BLOCK_THREADS (WAVES_PER_BLOCK * 32)
#define LDS_PITCH 33              // 16x33 f32 tile: pitch 33 -> conflict-free column reads (gcd(33,64)=1)

__global__ __launch_bounds__(BLOCK_THREADS)
void item_reg_wmma_kernel(const float* __restrict__ rating,
                          const int*   __restrict__ qtus,
                          const float* __restrict__ weight,
                          const float* __restrict__ b_user,
                          const float* __restrict__ b_item,
                          const int*   __restrict__ users,
                          const int*   __restrict__ items,
                          float*       __restrict__ out)
{
    // Per-wave product tile P[m][k] = w_jt * resid, m = output row, k = neighbor.
    __shared__ float P[WAVES_PER_BLOCK][OUTS_PER_WAVE][LDS_PITCH];

    const int lane = threadIdx.x & 31;
    const int wv   = threadIdx.x >> 5;
    const int b0   = (blockIdx.x * WAVES_PER_BLOCK + wv) * OUTS_PER_WAVE;

    // Lanes 0..15 preload per-output scalars for the wave's 16 outputs.
    int   u = 0, t = 0;
    float bu = 0.f, bt = 0.f;
    if (lane < OUTS_PER_WAVE) {
        u  = users[b0 + lane];
        t  = items[b0 + lane];
        bu = b_user[u];
        bt = b_item[t];
    }

    // Phase 1: gather. Iteration i = output row m; lane = neighbor k.
    // qtus row load is a coalesced 128B burst; weight/rating/b_item are L2 gathers.
    // All index math stays in 32-bit ints (max idx 2000*2000*32 < 2^31) so the
    // compiler can use SGPR-base + VGPR-offset (GVS) global loads.
    #pragma unroll 4
    for (int i = 0; i < OUTS_PER_WAVE; ++i) {
        const int   ui  = __shfl(u,  i, 32);
        const int   ti  = __shfl(t,  i, 32);
        const float bui = __shfl(bu, i, 32);

        const int   q  = qtus[(ui * I_DIM + ti) * K_DIM + lane];
        const float w  = weight[q * I_DIM + ti];
        const float bj = b_item[q];
        const float r  = rating[ui * I_DIM + q];

        P[wv][i][lane] = w * (r - bui - bj);
    }
    __syncthreads();

    // Phase 2: row-sum via V_WMMA_F32_16X16X4_F32 with an all-ones B matrix:
    //   D[m][n] = sum_k A[m,k] * 1  (permutation-invariant in k and in B layout).
    // A layout (32-bit A 16x4): lanes 0-15 hold K={0,1} in the 2 VGPRs, lanes
    // 16-31 hold K={2,3}; rows M = lane & 15. Eight chained WMMAs cover K=32.
    const int h = lane >> 4;      // half-wave selects K pair within chunk
    const int L = lane & 15;      // row M
    const v2f ones = {1.0f, 1.0f};
    v8f acc = {};
    #pragma unroll
    for (int c = 0; c < 8; ++c) {
        const int o = 4 * c + 2 * h;
        v2f a;
        a.x = P[wv][L][o];
        a.y = P[wv][L][o + 1];
        acc = __builtin_amdgcn_wmma_f32_16x16x4_f32(
            /*neg_a=*/false, a, /*neg_b=*/false, ones,
            /*c_mod=*/(short)0, acc, /*reuse_a=*/false, /*reuse_b=*/false);
    }

    // Diagonal extraction from the 16x16 f32 C/D layout:
    //   m < 8 : lane = m,      VGPR m
    //   m >= 8: lane = m + 16, VGPR m - 8
    const bool writer = (lane < 8) || (lane >= 24);
    const int  m      = lane & 15;
    const int  vidx   = m & 7;
    float d = acc[0];
    #pragma unroll
    for (int j = 1; j < 8; ++j) d = (vidx == j) ? acc[j] : d;

    const float bum = __shfl(bu, m, 32);
    const float btm = __shfl(bt, m, 32);
    if (writer) out[b0 + m] = bum + btm + d * (1.0f / 32.0f);
}

extern "C" void kernel_launch(void* const* d_in, const int* in_sizes, int n_in,
                              void* d_out, int out_size, void* d_ws, size_t ws_size,
                              hipStream_t stream) {
    const float* rating = (const float*)d_in[0];   // (2000, 2000) f32
    const int*   qtus   = (const int*)  d_in[1];   // (2000, 2000, 32) i32
    const float* weight = (const float*)d_in[2];   // (2000, 2000) f32
    const float* b_user = (const float*)d_in[3];   // (2000,) f32
    const float* b_item = (const float*)d_in[4];   // (2000,) f32
    const int*   users  = (const int*)  d_in[5];   // (16384,) i32
    const int*   items  = (const int*)  d_in[6];   // (16384,) i32
    float*       out    = (float*)d_out;           // (16384,) f32

    const int blocks = B_DIM / (WAVES_PER_BLOCK * OUTS_PER_WAVE); // 256 blocks x 128 threads
    item_reg_wmma_kernel<<<blocks, BLOCK_THREADS, 0, stream>>>(
        rating, qtus, weight, b_user, b_item, users, items, out);
}